// GCNModelAE_31439160606755
// MI455X (gfx1250) — compile-verified
//
#include <hip/hip_runtime.h>
#include <hip/hip_bf16.h>

// Problem constants (from reference)
#define NN  10000     // nodes (625 * 16)
#define EE  320000    // edges
#define FF  512       // input features
#define H1N 32        // hidden 1
#define H2N 16        // hidden 2 / z width
#define MT  (NN / 16) // 625 tiles

typedef __attribute__((ext_vector_type(2))) float v2f;
typedef __attribute__((ext_vector_type(8))) float v8f;

// ---------------------------------------------------------------------------
// zero scratch region (h1 .. z span)
// ---------------------------------------------------------------------------
__global__ void k_zero(float* __restrict__ p, int n) {
    int i = blockIdx.x * blockDim.x + threadIdx.x;
    if (i < n) p[i] = 0.0f;
}

// ---------------------------------------------------------------------------
// GEMM1: xw = x(10000x512) @ W1(512x32), fp32 WMMA 16x16x4, K = 128 steps
// one wave per 16x16 output tile; tiles = 625 * 2
// ---------------------------------------------------------------------------
__global__ void k_gemm1(const float* __restrict__ x,
                        const float* __restrict__ W1,
                        float* __restrict__ xw) {
    const int wave = threadIdx.x >> 5;
    const int tile = blockIdx.x * 8 + wave;
    if (tile >= MT * 2) return;                 // whole wave exits: EXEC stays all-1
    const int mt = tile >> 1, nt = tile & 1;
    const int lane = threadIdx.x & 31;
    const int m  = lane & 15;                   // M row (A) / N col (B,C,D)
    const int kh = lane >> 4;                   // K half select

    const float* __restrict__ arow = x  + (size_t)(mt * 16 + m) * FF;
    const float* __restrict__ bcol = W1 + nt * 16 + m;

    v8f c = {};
#pragma unroll 4
    for (int kk = 0; kk < FF / 4; ++kk) {
        const int k0 = kk * 4 + kh * 2;
        v2f a = *(const v2f*)(arow + k0);       // A[m][k0], A[m][k0+1]
        v2f b;                                  // B[k][n] = W1[k][n]
        b.x = bcol[(k0    ) * H1N];
        b.y = bcol[(k0 + 1) * H1N];
        c = __builtin_amdgcn_wmma_f32_16x16x4_f32(false, a, false, b,
                                                  (short)0, c, false, false);
    }
#pragma unroll
    for (int v = 0; v < 8; ++v)                 // D vgpr v: M = v (+8 for hi half)
        xw[(size_t)(mt * 16 + kh * 8 + v) * H1N + nt * 16 + m] = c[v];
}

// ---------------------------------------------------------------------------
// SpMM1 scatter: h1[row][f] += w * xw[col][f]   (f = 0..31), fp32 atomics
// one lane per (edge, feature) -> coalesced gather + coalesced atomic
// ---------------------------------------------------------------------------
__global__ void k_scatter1(const float* __restrict__ ew,
                           const int*  __restrict__ er,
                           const int*  __restrict__ ec,
                           const float* __restrict__ xw,
                           float* __restrict__ h1) {
    const int idx = blockIdx.x * 256 + threadIdx.x;   // 40000 * 256 == EE * 32
    const int e = idx >> 5, f = idx & 31;
    const float w = ew[e];
    const int r = er[e], cidx = ec[e];
    unsafeAtomicAdd(&h1[(size_t)r * H1N + f], w * xw[(size_t)cidx * H1N + f]);
}

// ---------------------------------------------------------------------------
// GEMM2: h2 = relu(h1)(10000x32) @ W2(32x16), K = 8 WMMA steps, ReLU fused on A
// ---------------------------------------------------------------------------
__global__ void k_gemm2(const float* __restrict__ h1,
                        const float* __restrict__ W2,
                        float* __restrict__ h2) {
    const int wave = threadIdx.x >> 5;
    const int mt = blockIdx.x * 8 + wave;
    if (mt >= MT) return;
    const int lane = threadIdx.x & 31;
    const int m  = lane & 15;
    const int kh = lane >> 4;

    const float* __restrict__ arow = h1 + (size_t)(mt * 16 + m) * H1N;
    const float* __restrict__ bcol = W2 + m;

    v8f c = {};
#pragma unroll
    for (int kk = 0; kk < H1N / 4; ++kk) {
        const int k0 = kk * 4 + kh * 2;
        v2f a;
        a.x = fmaxf(arow[k0    ], 0.0f);        // ReLU fused into A load
        a.y = fmaxf(arow[k0 + 1], 0.0f);
        v2f b;
        b.x = bcol[(k0    ) * H2N];
        b.y = bcol[(k0 + 1) * H2N];
        c = __builtin_amdgcn_wmma_f32_16x16x4_f32(false, a, false, b,
                                                  (short)0, c, false, false);
    }
#pragma unroll
    for (int v = 0; v < 8; ++v)
        h2[(size_t)(mt * 16 + kh * 8 + v) * H2N + m] = c[v];
}

// ---------------------------------------------------------------------------
// SpMM2 scatter: z[row][f] += w * h2[col][f]   (f = 0..15)
// ---------------------------------------------------------------------------
__global__ void k_scatter2(const float* __restrict__ ew,
                           const int*  __restrict__ er,
                           const int*  __restrict__ ec,
                           const float* __restrict__ h2,
                           float* __restrict__ z) {
    const int idx = blockIdx.x * 256 + threadIdx.x;   // 20000 * 256 == EE * 16
    const int e = idx >> 4, f = idx & 15;
    const float w = ew[e];
    const int r = er[e], cidx = ec[e];
    unsafeAtomicAdd(&z[(size_t)r * H2N + f], w * h2[(size_t)cidx * H2N + f]);
}

// ---------------------------------------------------------------------------
// Decoder: out = z @ z^T  (10000 x 10000), K = 16 -> 4 WMMA steps per tile.
// B = Zj^T so the B operand is loaded exactly like A from row block j.
// z (640 KB) stays resident in L2; output streamed with non-temporal stores.
// ---------------------------------------------------------------------------
__global__ void k_zzt(const float* __restrict__ z, float* __restrict__ out) {
    const int wave  = threadIdx.x >> 5;
    const int ntile = blockIdx.x * 8 + wave;
    if (ntile >= MT) return;
    const int mtile = blockIdx.y;
    const int lane = threadIdx.x & 31;
    const int m  = lane & 15;
    const int kh = lane >> 4;

    const float* __restrict__ arow = z + (size_t)(mtile * 16 + m) * H2N;
    const float* __restrict__ brow = z + (size_t)(ntile * 16 + m) * H2N;

    v8f c = {};
#pragma unroll
    for (int kk = 0; kk < 4; ++kk) {
        const int k0 = kk * 4 + kh * 2;
        v2f a = *(const v2f*)(arow + k0);
        v2f b = *(const v2f*)(brow + k0);
        c = __builtin_amdgcn_wmma_f32_16x16x4_f32(false, a, false, b,
                                                  (short)0, c, false, false);
    }
    const size_t base = (size_t)(mtile * 16 + kh * 8) * NN + (size_t)ntile * 16 + m;
#pragma unroll
    for (int v = 0; v < 8; ++v)
        __builtin_nontemporal_store(c[v], out + base + (size_t)v * NN);
}

// ---------------------------------------------------------------------------
extern "C" void kernel_launch(void* const* d_in, const int* in_sizes, int n_in,
                              void* d_out, int out_size, void* d_ws, size_t ws_size,
                              hipStream_t stream) {
    const float* x  = (const float*)d_in[0];
    const float* ew = (const float*)d_in[1];
    const float* W1 = (const float*)d_in[2];
    const float* W2 = (const float*)d_in[3];
    const int*   er = (const int*)d_in[4];
    const int*   ec = (const int*)d_in[5];
    float* out = (float*)d_out;

    // workspace layout (floats): xw[10000*32] | h1[10000*32] | h2[10000*16] | z[10000*16]
    float* xw = (float*)d_ws;
    float* h1 = xw + (size_t)NN * H1N;
    float* h2 = h1 + (size_t)NN * H1N;
    float* zz = h2 + (size_t)NN * H2N;

    // zero h1..z span (h2 included, harmless — overwritten by gemm2)
    const int zero_n = NN * H1N + 2 * NN * H2N;        // 640000 floats
    k_zero<<<(zero_n + 255) / 256, 256, 0, stream>>>(h1, zero_n);

    // GEMM1: 1250 tiles, 8 waves/block
    k_gemm1<<<(MT * 2 + 7) / 8, 256, 0, stream>>>(x, W1, xw);

    // SpMM1: E*32 lanes
    k_scatter1<<<(EE * 32) / 256, 256, 0, stream>>>(ew, er, ec, xw, h1);

    // GEMM2: 625 tiles
    k_gemm2<<<(MT + 7) / 8, 256, 0, stream>>>(h1, W2, h2);

    // SpMM2: E*16 lanes
    k_scatter2<<<(EE * 16) / 256, 256, 0, stream>>>(ew, er, ec, h2, zz);

    // Decoder: 625 x 625 tiles; grid.x covers n-tiles (8 waves/block), grid.y = m-tile
    dim3 g((MT + 7) / 8, MT);
    k_zzt<<<g, 256, 0, stream>>>(zz, out);
}